// ACT_67250597921543
// MI455X (gfx1250) — compile-verified
//
#include <hip/hip_runtime.h>
#include <math.h>

// ---------------------------------------------------------------------------
// ACT (adaptive computation time) for B=4, L=4096, D=512, F=2048, 11 steps.
// f32 matrix math on CDNA5 via V_WMMA_F32_16X16X4_F32 (wave32).
// Weights are repacked once per launch into WMMA B-fragment order so each
// lane's (k,k+1) pair is one coalesced 8-byte load.
// ---------------------------------------------------------------------------

typedef __attribute__((ext_vector_type(2))) float v2f;
typedef __attribute__((ext_vector_type(8))) float v8f;

#define DD   512          // model dim
#define FF   2048         // ffn dim
#define NTOK 16384        // B*L
#define NSTEP 11          // MAX_ITERATIONS + 1
#define ROWS 32           // token rows per workgroup (FFN kernel)
#define KF   64           // F-chunk width
#define SL   516          // s_lds row stride (512 + 4 pad -> conflict-free)
#define HL   68           // hid_lds row stride (64 + 4 pad)
#define THR  0.99f        // 1 - HALT_EPSILON

// ---------------------------------------------------------------------------
// repack W[K][N] (row major) -> WP[k4][h][n][j] = W[(k4*4 + h*2 + j)*N + n]
// so a B fragment for WMMA_16x16x4 is one v2f load per lane, coalesced.
// ---------------------------------------------------------------------------
__global__ __launch_bounds__(256) void act_repack(const float* __restrict__ W,
                                                  float* __restrict__ WP,
                                                  int K, int N) {
  int i = blockIdx.x * 256 + threadIdx.x;        // over K*N elements
  if (i >= K * N) return;
  int j  = i & 1;
  int t  = i >> 1;
  int n  = t % N;
  int t2 = t / N;
  int h  = t2 & 1;
  int k4 = t2 >> 1;
  WP[i] = W[(size_t)(k4 * 4 + h * 2 + j) * N + n];
}

// ---------------------------------------------------------------------------
// init: st <- state ; out(new_state) <- 0 ; hp/nup/rem <- 0
// ---------------------------------------------------------------------------
__global__ __launch_bounds__(256) void act_init(const float* __restrict__ state,
                                                float* __restrict__ st,
                                                float* __restrict__ out,
                                                float* __restrict__ hp) {
  int i = blockIdx.x * 256 + threadIdx.x;        // grid covers NTOK*DD
  st[i]  = state[i];
  out[i] = 0.f;
  if (i < NTOK) {
    hp[i] = 0.f;
    out[(size_t)NTOK * DD + i]        = 0.f;     // n_updates
    out[(size_t)NTOK * DD + NTOK + i] = 0.f;     // remainders
  }
}

// ---------------------------------------------------------------------------
// halting: one wave per token. p = sigmoid((st+emb)·Wp + bp); update hp/rem/
// nup and emit update weight uw for this step.
// ---------------------------------------------------------------------------
__global__ __launch_bounds__(256) void act_halt(const float* __restrict__ st,
                                                const float* __restrict__ emb,
                                                const float* __restrict__ Wp,
                                                const float* __restrict__ bp,
                                                float* __restrict__ hp,
                                                float* __restrict__ uw,
                                                float* __restrict__ nup,
                                                float* __restrict__ rem) {
  const int lane = threadIdx.x & 31;
  const int tok  = (blockIdx.x * 256 + threadIdx.x) >> 5;
  const float* s = st + (size_t)tok * DD;
  float acc = 0.f;
#pragma unroll 4
  for (int k = lane; k < DD; k += 32)
    acc = fmaf(s[k] + emb[k], Wp[k], acc);
  for (int off = 16; off > 0; off >>= 1)
    acc += __shfl_xor(acc, off, 32);
  if (lane == 0) {
    float p = 1.f / (1.f + expf(-(acc + bp[0])));
    float h = hp[tok], r = rem[tok], n = nup[tok];
    float still = (h < 1.f) ? 1.f : 0.f;
    float q   = h + p * still;
    float nh  = (q >  THR) ? still : 0.f;   // new_halted
    float st2 = (q <= THR) ? still : 0.f;   // still running
    h += p * st2;
    r += nh * (1.f - h);
    h += nh * r;
    n += st2 + nh;
    hp[tok]  = h;
    rem[tok] = r;
    nup[tok] = n;
    uw[tok]  = p * st2 + nh * r;
  }
}

// ---------------------------------------------------------------------------
// FFN: h = relu((st+emb) @ W1 + b1) @ W2 + b2 ; h *= mask ;
//      out += h * uw ; st = h      (all f32, WMMA 16x16x4 f32)
// W1p/W2p are in repacked fragment order (see act_repack).
// ---------------------------------------------------------------------------
__global__ __launch_bounds__(256) void act_ffn(float* __restrict__ st,
                                               const float* __restrict__ emb,
                                               const float* __restrict__ mask,
                                               const float* __restrict__ uw,
                                               const float* __restrict__ W1p,
                                               const float* __restrict__ b1,
                                               const float* __restrict__ W2p,
                                               const float* __restrict__ b2,
                                               float* __restrict__ out) {
  __shared__ __align__(16) float s_lds[ROWS * SL];
  __shared__ __align__(16) float hid_lds[ROWS * HL];
  __shared__ float mask_lds[ROWS];
  __shared__ float uw_lds[ROWS];

  const int  tid  = threadIdx.x;
  const int  lane = tid & 31;
  const int  wv   = tid >> 5;        // wave 0..7
  const int  r0   = blockIdx.x * ROWS;
  const bool hi   = lane >= 16;
  const int  ln   = lane & 15;
  const int  hsel = hi ? 1 : 0;

  // ---- stage s = st + step_emb[t] into LDS (32 x 512, padded stride)
  for (int i = tid; i < ROWS * (DD / 4); i += 256) {
    int row = i / (DD / 4);
    int c4  = (i % (DD / 4)) * 4;
    const float4 sv = *(const float4*)(st + (size_t)(r0 + row) * DD + c4);
    const float4 ev = *(const float4*)(emb + c4);
    s_lds[row * SL + c4 + 0] = sv.x + ev.x;
    s_lds[row * SL + c4 + 1] = sv.y + ev.y;
    s_lds[row * SL + c4 + 2] = sv.z + ev.z;
    s_lds[row * SL + c4 + 3] = sv.w + ev.w;
  }
  if (tid < ROWS) {
    mask_lds[tid] = mask[r0 + tid];
    uw_lds[tid]   = uw[r0 + tid];
  }
  __syncthreads();

  const int colD = wv * 64;                 // this wave's 64 output columns

  // output accumulators: acc[rowtile][coltile], seeded with b2 (same N per lane)
  v8f acc[2][4];
  for (int ct = 0; ct < 4; ++ct) {
    float bb = b2[colD + ct * 16 + ln];
    v8f v;
#pragma unroll
    for (int j = 0; j < 8; ++j) v[j] = bb;
    acc[0][ct] = v;
    acc[1][ct] = v;
  }

  const int rt1 = wv >> 2;                  // phase-1 row tile (0..1)
  const int ct1 = wv & 3;                   // phase-1 col tile (0..3)
  const int colF_loc = ct1 * 16 + ln;       // column within F-chunk

  for (int c = 0; c < FF; c += KF) {
    // ---------------- phase 1: hidden[32 x KF] = relu(s @ W1[:,chunk] + b1)
    v8f hacc = {};
    const int colF = c + colF_loc;
    {
      // slab base for (k4 = k>>2, half = hsel) at column colF
      const float* W1f = W1p + ((size_t)hsel * FF + colF) * 2;
#pragma unroll 4
      for (int k = 0; k < DD; k += 4) {
        const int kk = k + (hi ? 2 : 0);
        v2f a = *(const v2f*)(&s_lds[(rt1 * 16 + ln) * SL + kk]);
        v2f b = *(const v2f*)(W1f + (size_t)(k >> 2) * (2 * FF * 2));
        hacc = __builtin_amdgcn_wmma_f32_16x16x4_f32(
            false, a, false, b, (short)0, hacc, false, false);
      }
    }
    {   // bias + ReLU, scatter to padded LDS (C layout: M = j (+8 hi), N = ln)
      float bb = b1[colF];
#pragma unroll
      for (int j = 0; j < 8; ++j) {
        float hv = hacc[j] + bb;
        hv = hv > 0.f ? hv : 0.f;
        int row = rt1 * 16 + (hi ? j + 8 : j);
        hid_lds[row * HL + colF_loc] = hv;
      }
    }
    __syncthreads();

    // ---------------- phase 2: acc += hidden @ W2[chunk, colD:colD+64]
    const float* W2f = W2p +
        (((size_t)(c >> 2) * 2 + hsel) * DD + colD + ln) * 2;
#pragma unroll 2
    for (int k = 0; k < KF; k += 4) {
      const int kk = k + (hi ? 2 : 0);
      const float* W2k = W2f + (size_t)(k >> 2) * (2 * DD * 2);
      v2f bfrag[4];
#pragma unroll
      for (int ct = 0; ct < 4; ++ct)
        bfrag[ct] = *(const v2f*)(W2k + ct * 32);   // ct*16 cols * 2 floats
#pragma unroll
      for (int rt = 0; rt < 2; ++rt) {
        v2f a = *(const v2f*)(&hid_lds[(rt * 16 + ln) * HL + kk]);
#pragma unroll
        for (int ct = 0; ct < 4; ++ct) {
          acc[rt][ct] = __builtin_amdgcn_wmma_f32_16x16x4_f32(
              false, a, false, bfrag[ct], (short)0, acc[rt][ct], false, false);
        }
      }
    }
    __syncthreads();
  }

  // ---- epilogue: h = acc*mask ; out += h*uw ; st = h (rows WG-disjoint)
  for (int rt = 0; rt < 2; ++rt) {
#pragma unroll
    for (int ct = 0; ct < 4; ++ct) {
      int col = colD + ct * 16 + ln;
#pragma unroll
      for (int j = 0; j < 8; ++j) {
        int lrow = rt * 16 + (hi ? j + 8 : j);
        size_t idx = (size_t)(r0 + lrow) * DD + col;
        float h = acc[rt][ct][j] * mask_lds[lrow];
        out[idx] += h * uw_lds[lrow];
        st[idx] = h;
      }
    }
  }
}

// ---------------------------------------------------------------------------
extern "C" void kernel_launch(void* const* d_in, const int* in_sizes, int n_in,
                              void* d_out, int out_size, void* d_ws,
                              size_t ws_size, hipStream_t stream) {
  (void)in_sizes; (void)n_in; (void)out_size; (void)ws_size;
  const float* state = (const float*)d_in[0];
  const float* mask  = (const float*)d_in[1];
  const float* emb   = (const float*)d_in[2];   // step_emb [11, 512]
  const float* Wp    = (const float*)d_in[3];
  const float* bp    = (const float*)d_in[4];
  const float* W1    = (const float*)d_in[5];
  const float* b1    = (const float*)d_in[6];
  const float* W2    = (const float*)d_in[7];
  const float* b2    = (const float*)d_in[8];

  float* out = (float*)d_out;                    // [NTOK*DD | NTOK | NTOK]
  float* nup = out + (size_t)NTOK * DD;
  float* rem = nup + NTOK;

  float* st  = (float*)d_ws;                     // NTOK*DD carried state
  float* hp  = st + (size_t)NTOK * DD;           // NTOK halting prob
  float* uw  = hp + NTOK;                        // NTOK update weights
  float* W1p = uw + NTOK;                        // DD*FF repacked W1
  float* W2p = W1p + (size_t)DD * FF;            // FF*DD repacked W2

  // one-time (per launch) weight repack into WMMA fragment order
  act_repack<<<dim3((DD * FF + 255) / 256), dim3(256), 0, stream>>>(
      W1, W1p, DD, FF);
  act_repack<<<dim3((FF * DD + 255) / 256), dim3(256), 0, stream>>>(
      W2, W2p, FF, DD);

  act_init<<<dim3((NTOK * DD) / 256), dim3(256), 0, stream>>>(state, st, out, hp);

  for (int t = 0; t < NSTEP; ++t) {
    const float* embt = emb + (size_t)t * DD;
    act_halt<<<dim3(NTOK * 32 / 256), dim3(256), 0, stream>>>(
        st, embt, Wp, bp, hp, uw, nup, rem);
    act_ffn<<<dim3(NTOK / ROWS), dim3(256), 0, stream>>>(
        st, embt, mask, uw, W1p, b1, W2p, b2, out);
  }
}